// GNNModel_83107617177903
// MI455X (gfx1250) — compile-verified
//
#include <hip/hip_runtime.h>

typedef float v2f __attribute__((ext_vector_type(2)));
typedef float v8f __attribute__((ext_vector_type(8)));

#define FEAT   256
#define HIDDEN 256
#define EMBED  128

// ---------------- small utility kernels ----------------

__global__ void fill_i32(int* __restrict__ p, int v, int n) {
    int i = blockIdx.x * blockDim.x + threadIdx.x;
    if (i < n) p[i] = v;
}

__global__ void copy_i32(const int* __restrict__ a, int* __restrict__ b, int n) {
    int i = blockIdx.x * blockDim.x + threadIdx.x;
    if (i < n) b[i] = a[i];
}

__global__ void count_edges(const int* __restrict__ dst, int* __restrict__ cnt, int E) {
    int i = blockIdx.x * blockDim.x + threadIdx.x;
    if (i < E) atomicAdd(&cnt[dst[i]], 1);
}

// dis = rsqrt(in-degree + 1 self-loop); deg >= 1 always
__global__ void dis_from_cnt(const int* __restrict__ cnt, float* __restrict__ dis, int n) {
    int i = blockIdx.x * blockDim.x + threadIdx.x;
    if (i < n) dis[i] = rsqrtf((float)cnt[i] + 1.0f);
}

// Single-block exclusive scan of cnt[0..n) -> rowptr[0..n]; rowptr[n] = total.
__global__ __launch_bounds__(1024) void scan_rowptr(const int* __restrict__ cnt,
                                                    int* __restrict__ rowptr, int n) {
    __shared__ int part[1024];
    __shared__ int base[1024];
    int t = threadIdx.x;
    int per = (n + 1023) >> 10;
    int s0 = t * per;
    int e0 = min(s0 + per, n);
    int sum = 0;
    for (int i = s0; i < e0; ++i) sum += cnt[i];
    part[t] = sum;
    __syncthreads();
    if (t == 0) {
        int run = 0;
        for (int i = 0; i < 1024; ++i) { base[i] = run; run += part[i]; }
        rowptr[n] = run;   // == E
    }
    __syncthreads();
    int run = base[t];
    for (int i = s0; i < e0; ++i) { rowptr[i] = run; run += cnt[i]; }
}

// Bucket edges by destination: ecol[pos] = src, enrm[pos] = dis[src]*dis[dst]
__global__ void fill_edges(const int* __restrict__ src, const int* __restrict__ dst,
                           const float* __restrict__ dis, int* __restrict__ cursor,
                           int* __restrict__ ecol, float* __restrict__ enrm, int E) {
    int e = blockIdx.x * blockDim.x + threadIdx.x;
    if (e >= E) return;
    int s = src[e], d = dst[e];
    int pos = atomicAdd(&cursor[d], 1);
    ecol[pos] = s;
    enrm[pos] = dis[s] * dis[d];
}

// ---------------- fp32 WMMA GEMM: C[M,N] = A[M,256] x B[256,N] ----------------
// One wave -> one 16x16 C tile. Block = 8 waves = 128 rows, one shared N-tile.
// K fixed at 256 (both layers). B slab staged in LDS (256x16 f32 = 16KB).

__global__ __launch_bounds__(256) void gemm_f32_wmma(const float* __restrict__ A,
                                                     const float* __restrict__ B,
                                                     float* __restrict__ C,
                                                     int M, int N) {
    __shared__ float sB[256 * 16];   // sB[k*16 + n]

    const int ntile = blockIdx.x * 16;
    for (int j = threadIdx.x; j < 256 * 16; j += 256) {
        int k = j >> 4;
        int n = j & 15;
        sB[j] = B[k * N + ntile + n];
    }
    __syncthreads();

    const int wave = threadIdx.x >> 5;
    const int lane = threadIdx.x & 31;
    const int rowBase = (blockIdx.y * 8 + wave) * 16;
    if (rowBase >= M) return;   // wave-uniform: EXEC stays all-1s for WMMA

    // A 16x4 f32 layout: lanes 0-15 row M=lane, VGPR0/1 = K,K+1;
    // lanes 16-31 same rows, K+2,K+3.  -> aligned float2 per lane.
    const int mrow = rowBase + (lane & 15);
    const int koff = (lane >> 4) * 2;
    const float* __restrict__ arow = A + (size_t)mrow * 256 + koff;
    // B 4x16 f32 layout: col = lane&15; lanes 0-15 K, lanes 16-31 K+2.
    const float* __restrict__ bcol = sB + koff * 16 + (lane & 15);

    v8f acc = {};
#pragma unroll 8
    for (int k = 0; k < 256; k += 4) {
        v2f a = *(const v2f*)(arow + k);
        v2f b;
        b.x = bcol[k * 16];
        b.y = bcol[k * 16 + 16];
        acc = __builtin_amdgcn_wmma_f32_16x16x4_f32(
            false, a, false, b, (short)0, acc, false, false);
    }

    // C/D 16x16 f32 layout: VGPR r -> M = r + 8*(lane>>4), N = lane&15.
    const int crow = rowBase + (lane >> 4) * 8;
    const int ccol = ntile + (lane & 15);
#pragma unroll
    for (int r = 0; r < 8; ++r)
        C[(size_t)(crow + r) * N + ccol] = acc[r];
}

// ---------------- CSR gather aggregation (atomic-free accumulation) ----------------
// Whole waves own one destination node: edge-list loads are wave-uniform
// (broadcast), feature gathers are coalesced float4 bursts. Accumulates
// bias + self-loop + all incoming messages in registers, one store at end.

template <int F, bool RELU>
__global__ __launch_bounds__(256) void gather_agg(const float* __restrict__ xw,
                                                  const float* __restrict__ dis,
                                                  const float* __restrict__ bias,
                                                  const int* __restrict__ rowptr,
                                                  const int* __restrict__ ecol,
                                                  const float* __restrict__ enrm,
                                                  float* __restrict__ out, int n) {
    constexpr int C = F / 4;        // float4 chunks per node
    constexpr int WPN = C / 32;     // waves per node (256->2, 128->1)
    int w = blockIdx.x * 8 + (threadIdx.x >> 5);
    int lane = threadIdx.x & 31;
    int node = w / WPN;
    int c = (w % WPN) * 32 + lane;
    if (node >= n) return;

    const float4* x4 = (const float4*)xw;
    float dv = dis[node];
    float s2 = dv * dv;
    float4 acc = ((const float4*)bias)[c];
    float4 v = x4[(size_t)node * C + c];
    acc.x += v.x * s2; acc.y += v.y * s2; acc.z += v.z * s2; acc.w += v.w * s2;

    int beg = rowptr[node];
    int end = rowptr[node + 1];
    for (int j = beg; j < end; ++j) {
        int s = ecol[j];          // wave-uniform
        float wgt = enrm[j];      // wave-uniform
        float4 m = x4[(size_t)s * C + c];
        acc.x += m.x * wgt; acc.y += m.y * wgt; acc.z += m.z * wgt; acc.w += m.w * wgt;
    }
    if (RELU) {
        acc.x = fmaxf(acc.x, 0.0f); acc.y = fmaxf(acc.y, 0.0f);
        acc.z = fmaxf(acc.z, 0.0f); acc.w = fmaxf(acc.w, 0.0f);
    }
    ((float4*)out)[(size_t)node * C + c] = acc;
}

// ---------------- launch ----------------

static inline size_t alignup(size_t x) { return (x + 255) & ~(size_t)255; }

extern "C" void kernel_launch(void* const* d_in, const int* in_sizes, int n_in,
                              void* d_out, int out_size, void* d_ws, size_t ws_size,
                              hipStream_t stream) {
    const float* x  = (const float*)d_in[0];
    const int*   ei = (const int*)d_in[1];   // [2, E]
    const float* W1 = (const float*)d_in[2];
    const float* b1 = (const float*)d_in[3];
    const float* W2 = (const float*)d_in[4];
    const float* b2 = (const float*)d_in[5];

    const int N = in_sizes[0] / FEAT;   // 50000
    const int E = in_sizes[1] / 2;      // 800000
    const int* src = ei;
    const int* dst = ei + E;

    char* p = (char*)d_ws;
    float* dis    = (float*)p; p += alignup((size_t)N * 4);
    int*   rowptr = (int*)p;   p += alignup((size_t)(N + 1) * 4);
    int*   cursor = (int*)p;   p += alignup((size_t)N * 4);   // counts, then fill cursors
    int*   ecol   = (int*)p;   p += alignup((size_t)E * 4);
    float* enrm   = (float*)p; p += alignup((size_t)E * 4);
    float* xw1    = (float*)p; p += (size_t)N * HIDDEN * 4;
    float* h      = (float*)p; p += (size_t)N * HIDDEN * 4;
    float* hw2    = (float*)p;
    float* out    = (float*)d_out;

    const int T = 256;

    // ---- CSR build + normalization ----
    fill_i32<<<(N + T - 1) / T, T, 0, stream>>>(cursor, 0, N);
    count_edges<<<(E + T - 1) / T, T, 0, stream>>>(dst, cursor, E);
    dis_from_cnt<<<(N + T - 1) / T, T, 0, stream>>>(cursor, dis, N);
    scan_rowptr<<<1, 1024, 0, stream>>>(cursor, rowptr, N);
    copy_i32<<<(N + T - 1) / T, T, 0, stream>>>(rowptr, cursor, N);
    fill_edges<<<(E + T - 1) / T, T, 0, stream>>>(src, dst, dis, cursor, ecol, enrm, E);

    // ---- layer 1: xw1 = x @ W1 ; h = relu(gather + self + b1) ----
    gemm_f32_wmma<<<dim3(HIDDEN / 16, (N + 127) / 128), T, 0, stream>>>(x, W1, xw1, N, HIDDEN);
    {
        constexpr int WPN = (HIDDEN / 4) / 32;   // 2
        int blocks = (N * WPN + 7) / 8;
        gather_agg<HIDDEN, true><<<blocks, T, 0, stream>>>(xw1, dis, b1, rowptr, ecol, enrm, h, N);
    }

    // ---- layer 2: hw2 = h @ W2 ; out = gather + self + b2 ----
    gemm_f32_wmma<<<dim3(EMBED / 16, (N + 127) / 128), T, 0, stream>>>(h, W2, hw2, N, EMBED);
    {
        constexpr int WPN = (EMBED / 4) / 32;    // 1
        int blocks = (N * WPN + 7) / 8;
        gather_agg<EMBED, false><<<blocks, T, 0, stream>>>(hw2, dis, b2, rowptr, ecol, enrm, out, N);
    }
}